// MinLoss_12343736009330
// MI455X (gfx1250) — compile-verified
//
#include <hip/hip_runtime.h>
#include <math.h>

typedef __attribute__((ext_vector_type(2))) float v2f;
typedef __attribute__((ext_vector_type(8))) float v8f;

#define NS    4
#define SEQ   512
#define BSZ   16
#define DIM   512
#define LFLAT (SEQ * DIM)          /* 262144 per-source flattened length */
#define BD    (BSZ * DIM)          /* 8192  */
#define TBD   (SEQ * BSZ * DIM)    /* preds stride per source           */
#define GX    64                   /* K-chunks per batch-group          */
#define WAVES 8
#define KPERBLOCK (LFLAT / GX)         /* 4096 */
#define KPERWAVE  (KPERBLOCK / WAVES)  /* 512  */
#define REC   288                  /* per-block record: 256 dots + 16 pn + 16 gn */

// Each wave's K-slice must stay inside one D-row of preds so both streams are
// contiguous (k0 is a multiple of KPERWAVE; KPERWAVE divides DIM).
static_assert(KPERWAVE <= DIM && (DIM % KPERWAVE) == 0, "pred stream not contiguous");
static_assert((KPERWAVE % 4) == 0, "K step is 4");

// Stage 1: per (batch-group, K-chunk) block, accumulate the 16x16 dot-product
// tile (4 batches x 4 sources on both M and N; only same-batch entries, m%4 == n%4, are
// consumed later) with V_WMMA_F32_16X16X4_F32, plus per-row ||p||^2 and
// per-col ||g||^2 partials. Deterministic wave->block->workspace reduction.
__global__ __launch_bounds__(256) void minloss_partial(
    const float* __restrict__ preds,   // [S, T, B, D]
    const float* __restrict__ gts,     // [S, B, T, D]
    float* __restrict__ ws) {
  __shared__ float sh[WAVES * 320];

  const int tid    = threadIdx.x;
  const int w      = tid >> 5;
  const int lane   = tid & 31;
  const int laneHi = lane >> 4;      // upper half-wave handles K+2,K+3
  const int r      = lane & 15;      // A row index == B col index (m/n)
  const int s      = r >> 2;         // source
  const int bi     = r & 3;          // batch within group
  const int bg     = blockIdx.y;
  const int b      = bg * 4 + bi;    // global batch

  // preds[s, t, b, d]  -> base + t*B*D + d
  const size_t p_base = (size_t)s * TBD + (size_t)b * DIM;
  // gts[s, b, t, d]    -> fully contiguous in l = t*D + d
  const size_t g_base = ((size_t)s * BSZ + (size_t)b) * (size_t)LFLAT;

  const int k0 = blockIdx.x * KPERBLOCK + w * KPERWAVE;  // multiple of KPERWAVE

  // Whole slice lies in row t0 = k0/DIM of preds -> both streams contiguous.
  const float* __restrict__ pptr =
      preds + p_base + (size_t)(k0 >> 9) * BD + (size_t)(k0 & (DIM - 1)) + 2 * laneHi;
  const float* __restrict__ gptr =
      gts + g_base + (size_t)k0 + 2 * laneHi;

  v8f  c  = {};
  float pn = 0.0f, gn = 0.0f;

#pragma unroll 4
  for (int i = 0; i < KPERWAVE / 4; ++i) {
    v2f a  = *(const v2f*)(pptr + (size_t)i * 4);
    v2f bb = *(const v2f*)(gptr + (size_t)i * 4);

    // D = A(16x4) * B(4x16) + C : accumulates all 16x16 pairwise dot partials
    c = __builtin_amdgcn_wmma_f32_16x16x4_f32(
        /*neg_a=*/false, a, /*neg_b=*/false, bb,
        /*c_mod=*/(short)0, c, /*reuse_a=*/false, /*reuse_b=*/false);

    pn = fmaf(a.x,  a.x,  pn); pn = fmaf(a.y,  a.y,  pn);
    gn = fmaf(bb.x, bb.x, gn); gn = fmaf(bb.y, bb.y, gn);
  }

  // Stash per-wave results (C layout: VGPR v, lanes<16 -> M=v, lanes>=16 -> M=v+8, N=lane%16)
  float* shw = sh + w * 320;
#pragma unroll
  for (int v = 0; v < 8; ++v) {
    const int m = v + laneHi * 8;
    shw[m * 16 + r] = c[v];
  }
  shw[256 + lane] = pn;   // row r partial (two lanes per row: lane, lane+16)
  shw[288 + lane] = gn;   // col r partial
  __syncthreads();

  // Fixed-order cross-wave reduction -> per-block workspace record
  float* rec = ws + ((size_t)bg * GX + (size_t)blockIdx.x) * REC;
  {
    float sum = 0.0f;
#pragma unroll
    for (int ww = 0; ww < WAVES; ++ww) sum += sh[ww * 320 + tid];
    rec[tid] = sum;                         // 256 dot entries
  }
  if (tid < 16) {
    float ps = 0.0f, gs = 0.0f;
#pragma unroll
    for (int ww = 0; ww < WAVES; ++ww) {
      ps += sh[ww * 320 + 256 + tid] + sh[ww * 320 + 256 + 16 + tid];
      gs += sh[ww * 320 + 288 + tid] + sh[ww * 320 + 288 + 16 + tid];
    }
    rec[256 + tid] = ps;
    rec[272 + tid] = gs;
  }
}

// Stage 2: fixed-order sum over K-chunks, build 4x4 distance matrices,
// greedy global-argmin matching (row-major first-min tie-break, as jnp.argmin),
// emit the scalar loss.
__global__ __launch_bounds__(256) void minloss_final(
    const float* __restrict__ ws, float* __restrict__ out) {
  __shared__ float dsh[4 * 256];
  __shared__ float pnsh[4 * 16];
  __shared__ float gnsh[4 * 16];
  const int t = threadIdx.x;

  for (int bg = 0; bg < 4; ++bg) {
    const float* base = ws + (size_t)bg * GX * REC;
    float s = 0.0f;
    for (int gx = 0; gx < GX; ++gx) s += base[(size_t)gx * REC + t];
    dsh[bg * 256 + t] = s;
    if (t < 16) {
      float ps = 0.0f, gs = 0.0f;
      for (int gx = 0; gx < GX; ++gx) {
        ps += base[(size_t)gx * REC + 256 + t];
        gs += base[(size_t)gx * REC + 272 + t];
      }
      pnsh[bg * 16 + t] = ps;
      gnsh[bg * 16 + t] = gs;
    }
  }
  __syncthreads();

  if (t == 0) {
    float total = 0.0f;
    for (int bg = 0; bg < 4; ++bg) {
      for (int bi = 0; bi < 4; ++bi) {
        float dist[16];
        for (int s1 = 0; s1 < 4; ++s1) {
          for (int s2 = 0; s2 < 4; ++s2) {
            const int m = s1 * 4 + bi, n = s2 * 4 + bi;
            float d2 = pnsh[bg * 16 + m] + gnsh[bg * 16 + n]
                     - 2.0f * dsh[bg * 256 + m * 16 + n];
            dist[s1 * 4 + s2] = sqrtf(fmaxf(d2, 0.0f));
          }
        }
        bool rUsed[4] = {false, false, false, false};
        bool cUsed[4] = {false, false, false, false};
        for (int it = 0; it < 4; ++it) {
          float best = INFINITY; int br = 0, bc = 0;
          for (int rr = 0; rr < 4; ++rr) {
            for (int cc = 0; cc < 4; ++cc) {
              if (rUsed[rr] || cUsed[cc]) continue;
              const float v = dist[rr * 4 + cc];
              if (v < best) { best = v; br = rr; bc = cc; }
            }
          }
          rUsed[br] = true; cUsed[bc] = true;
          total += best;   // ||p_s - g_match(s)|| == selected dist entry
        }
      }
    }
    out[0] = total;
  }
}

extern "C" void kernel_launch(void* const* d_in, const int* in_sizes, int n_in,
                              void* d_out, int out_size, void* d_ws, size_t ws_size,
                              hipStream_t stream) {
  const float* preds = (const float*)d_in[0];
  const float* gts   = (const float*)d_in[1];
  float* ws = (float*)d_ws;   // needs 4*GX*REC*4 = 294,912 bytes

  dim3 grid(GX, 4);
  minloss_partial<<<grid, 256, 0, stream>>>(preds, gts, ws);
  minloss_final<<<1, 256, 0, stream>>>(ws, (float*)d_out);
}